// SelfAttention_5145370820771
// MI455X (gfx1250) — compile-verified
//
#include <hip/hip_runtime.h>

typedef __attribute__((ext_vector_type(2))) float v2f;
typedef __attribute__((ext_vector_type(8))) float v8f;

#define LOG2E 1.4426950408889634f
#define NBATCH 8
#define NPIX   4096   // 64*64
#define NCH    8

// device-safe fast transcendentals (base-2, map to v_exp_f32 / v_log_f32)
#define FAST_EXP2(x) __builtin_amdgcn_exp2f(x)
#define FAST_LOG2(x) __builtin_amdgcn_logf(x)

// ---------------- Kernel 1: q/k/v 1x1-conv projections (d=1) ----------------
__global__ __launch_bounds__(256) void qkv_proj(
    const float* __restrict__ x,
    const float* __restrict__ Wq, const float* __restrict__ bq,
    const float* __restrict__ Wk, const float* __restrict__ bk,
    const float* __restrict__ Wv, const float* __restrict__ bv,
    float* __restrict__ q, float* __restrict__ k, float* __restrict__ v)
{
    int t = blockIdx.x * 256 + threadIdx.x;          // 0 .. B*N-1
    const float4* xp = (const float4*)(x + (size_t)t * NCH);
    float4 x0 = xp[0];
    float4 x1 = xp[1];
    float xa[8] = {x0.x, x0.y, x0.z, x0.w, x1.x, x1.y, x1.z, x1.w};
    float qa = bq[0], ka = bk[0], va = bv[0];
#pragma unroll
    for (int c = 0; c < 8; ++c) {
        qa = fmaf(xa[c], Wq[c], qa);
        ka = fmaf(xa[c], Wk[c], ka);
        va = fmaf(xa[c], Wv[c], va);
    }
    q[t] = qa; k[t] = ka; v[t] = va;
}

// ------ Kernel 2: per-row softmax stats (m, log2 denom) + y = gamma*out + x ------
// grid: B*64 blocks (64 rows/block), 256 threads = 64 rows x 4 j-slices.
__global__ __launch_bounds__(256) void softmax_stats(
    const float* __restrict__ q, const float* __restrict__ k, const float* __restrict__ v,
    const float* __restrict__ x, const float* __restrict__ gamma,
    float* __restrict__ c, float* __restrict__ y)
{
    const int b   = blockIdx.x >> 6;
    const int i0  = (blockIdx.x & 63) * 64;
    const int tid = threadIdx.x;

    // 4 slices of 1024 floats, padded by 4 floats to de-conflict LDS banks
    __shared__ float sK[4 * 1028];
    __shared__ float sV[4 * 1028];
    __shared__ float red[512];

    const float* kb = k + b * NPIX;
    const float* vb = v + b * NPIX;
    for (int t = tid; t < NPIX; t += 256) {
        int idx = (t >> 10) * 1028 + (t & 1023);
        sK[idx] = kb[t];
        sV[idx] = vb[t];
    }
    __syncthreads();

    // block-wide kmax/kmin (rank-1 softmax max trick)
    float mx = -3.402823466e38f, mn = 3.402823466e38f;
    for (int t = tid; t < NPIX; t += 256) {
        float kv = sK[(t >> 10) * 1028 + (t & 1023)];
        mx = fmaxf(mx, kv);
        mn = fminf(mn, kv);
    }
    red[tid] = mx; red[256 + tid] = mn;
    __syncthreads();
    for (int s = 128; s > 0; s >>= 1) {
        if (tid < s) {
            red[tid]       = fmaxf(red[tid], red[tid + s]);
            red[256 + tid] = fminf(red[256 + tid], red[256 + tid + s]);
        }
        __syncthreads();
    }
    const float kmax = red[0], kmin = red[256];

    const int row   = tid >> 2;   // 0..63
    const int slice = tid & 3;    // 0..3  (j-range slice*1024 .. +1024)
    const int i     = i0 + row;

    const float qi = q[b * NPIX + i];
    const float q2 = qi * LOG2E;
    const float m2 = (q2 >= 0.0f) ? q2 * kmax : q2 * kmin;   // = max_j(q_i*k_j)*log2e

    const float4* kp = (const float4*)(sK + slice * 1028);
    const float4* vp = (const float4*)(sV + slice * 1028);
    float denom = 0.0f, num = 0.0f;
#pragma unroll 4
    for (int jj = 0; jj < 256; ++jj) {
        float4 k4 = kp[jj];
        float4 v4 = vp[jj];
        float e0 = FAST_EXP2(fmaf(q2, k4.x, -m2));
        float e1 = FAST_EXP2(fmaf(q2, k4.y, -m2));
        float e2 = FAST_EXP2(fmaf(q2, k4.z, -m2));
        float e3 = FAST_EXP2(fmaf(q2, k4.w, -m2));
        denom += (e0 + e1) + (e2 + e3);
        num = fmaf(e0, v4.x, num);
        num = fmaf(e1, v4.y, num);
        num = fmaf(e2, v4.z, num);
        num = fmaf(e3, v4.w, num);
    }
    // combine the 4 j-slices (adjacent lanes, wave32 shuffles)
    denom += __shfl_xor(denom, 1);
    num   += __shfl_xor(num,   1);
    denom += __shfl_xor(denom, 2);
    num   += __shfl_xor(num,   2);

    if (slice == 0) {
        c[b * NPIX + i] = m2 + FAST_LOG2(denom);   // att = exp2(q2*k_j - c)
        float outv = num / denom;
        float g = gamma[0];
        const float4* xp = (const float4*)(x + ((size_t)b * NPIX + i) * NCH);
        float4 x0 = xp[0], x1 = xp[1];
        float4 y0, y1;
        y0.x = fmaf(outv, g, x0.x); y0.y = fmaf(outv, g, x0.y);
        y0.z = fmaf(outv, g, x0.z); y0.w = fmaf(outv, g, x0.w);
        y1.x = fmaf(outv, g, x1.x); y1.y = fmaf(outv, g, x1.y);
        y1.z = fmaf(outv, g, x1.z); y1.w = fmaf(outv, g, x1.w);
        float4* yp = (float4*)(y + ((size_t)b * NPIX + i) * NCH);
        yp[0] = y0; yp[1] = y1;
    }
}

// ------ Kernel 3: stream the 537MB attention matrix via WMMA outer-product tiles ------
// grid: (B*32 row-blocks of 128 rows, 4 j-slices of 1024 cols), 256 threads = 8 waves.
// Each wave: 16 rows, one 16x16 tile per v_wmma_f32_16x16x4_f32, exp2, NT store.
// All loads are non-divergent (full EXEC throughout; required for WMMA and avoids
// per-iteration exec save/restore).
__global__ __launch_bounds__(256) void attn_write(
    const float* __restrict__ q, const float* __restrict__ k,
    const float* __restrict__ c, float* __restrict__ attn)
{
    const int b      = blockIdx.x >> 5;
    const int i_base = (blockIdx.x & 31) * 128;
    const int j_base = blockIdx.y * 1024;
    const int tid    = threadIdx.x;

    __shared__ float sK[1024];
    for (int t = tid; t < 1024; t += 256)
        sK[t] = k[b * NPIX + j_base + t];
    __syncthreads();

    const int wave = tid >> 5;
    const int lane = tid & 31;
    const int lo16 = lane & 15;          // always-in-range index
    const bool low = (lane < 16);
    const int i0   = i_base + wave * 16;

    // A (16x4 f32): column K=0 = q rows, everything else 0. Uniform load + select
    // (no EXEC divergence).
    float qraw = q[b * NPIX + i0 + lo16];
    v2f A; A[0] = low ? qraw : 0.0f; A[1] = 0.0f;

    // per-lane softmax constants for its 8 output rows (M = r + 8*(lane>=16))
    const int rbase = i0 + ((lane >> 4) << 3);
    float cc[8];
#pragma unroll
    for (int r = 0; r < 8; ++r)
        cc[r] = c[b * NPIX + rbase + r];

    float* outp = attn + (((size_t)b * NPIX + rbase) * NPIX + j_base + lo16);

    for (int jt = 0; jt < 1024; jt += 16) {
        // B (4x16 f32): row K=0 = k cols, everything else 0.
        // Non-divergent LDS read (lanes 16-31 read a valid dup, then masked to 0).
        float kraw = sK[jt + lo16];
        v2f Bm; Bm[0] = low ? kraw : 0.0f; Bm[1] = 0.0f;
        v8f Cz = {0.0f, 0.0f, 0.0f, 0.0f, 0.0f, 0.0f, 0.0f, 0.0f};
        // energy tile = q (x) k  (rank-1; only K=0 contributes)
        v8f D = __builtin_amdgcn_wmma_f32_16x16x4_f32(
            false, A, false, Bm, (short)0, Cz, false, false);
#pragma unroll
        for (int r = 0; r < 8; ++r) {
            float att = FAST_EXP2(fmaf(D[r], LOG2E, -cc[r]));
            __builtin_nontemporal_store(att, outp + (size_t)r * NPIX + jt);
        }
    }
}

extern "C" void kernel_launch(void* const* d_in, const int* in_sizes, int n_in,
                              void* d_out, int out_size, void* d_ws, size_t ws_size,
                              hipStream_t stream) {
    const float* x     = (const float*)d_in[0];
    const float* Wq    = (const float*)d_in[1];
    const float* bq    = (const float*)d_in[2];
    const float* Wk    = (const float*)d_in[3];
    const float* bk    = (const float*)d_in[4];
    const float* Wv    = (const float*)d_in[5];
    const float* bv    = (const float*)d_in[6];
    const float* gamma = (const float*)d_in[7];

    float* y    = (float*)d_out;                       // (8,64,64,8) = 262144 f32
    float* attn = (float*)d_out + NBATCH * NPIX * NCH; // (8,4096,4096) f32

    float* ws = (float*)d_ws;                          // 512 KB used
    float* q  = ws;
    float* k  = ws + NBATCH * NPIX;
    float* v  = ws + 2 * NBATCH * NPIX;
    float* c  = ws + 3 * NBATCH * NPIX;

    qkv_proj<<<(NBATCH * NPIX) / 256, 256, 0, stream>>>(x, Wq, bq, Wk, bk, Wv, bv, q, k, v);
    softmax_stats<<<NBATCH * 64, 256, 0, stream>>>(q, k, v, x, gamma, c, y);
    attn_write<<<dim3(NBATCH * 32, 4), 256, 0, stream>>>(q, k, c, attn);
}